// CRF_87428354278098
// MI455X (gfx1250) — compile-verified
//
#include <hip/hip_runtime.h>
#include <stdint.h>

// ---------------------------------------------------------------------------
// Viterbi decode (CRF), T=65536 steps, K=512 tags, on gfx1250 (MI455X).
//
// Forward: 16 persistent WGs x 1024 threads (32 wave32s). The max-plus step
//   fv'[n] = max_p (fv[p] + trans[n,p]) + feat[t,n]
// is computed as WMMA broadcast-add + VALU argmax:
//   D = A x B + C  with  C = trans^T tile (VGPR-resident),
//   A col0 = fv[prev], B = ones  =>  D[p,n] = fv[p] + trans[n,p]  (exact).
// D's layout puts prev across VGPRs and next across lanes, so the argmax over
// prev is per-lane (no shuffles) + one half-merge + a 16-way segment combine.
// fv is exchanged through L2 with a global spin barrier; feats are prefetched
// via CDNA5 async-to-LDS (global_load_async_to_lds_b32 / s_wait_asynccnt).
// Backtrack: chunked parallel function composition over the u16 backpointer
// table (64MB, fully resident in the 192MB L2).
// ---------------------------------------------------------------------------

#define TT 65536
#define KK 512
#define NEGV (-10000.0f)
#define NWG 16          // persistent workgroups (one per WGP)
#define RPW 32          // "next" rows per WG  (NWG*RPW == KK)
#define NTHREADS 1024   // 32 wave32s per WG
#define PF 6            // feats prefetch depth (async pipeline)
#define FR 8            // feats LDS ring slots (power of two > PF)
#define LCH 256         // backtrack chunk length
#define CCH (TT / LCH)  // 256 chunks

typedef __attribute__((address_space(3))) const void lds_cv;
typedef __attribute__((ext_vector_type(2))) float v2f;
typedef __attribute__((ext_vector_type(8))) float v8f;

// Raw LDS byte offset of a __shared__ object (addrspacecast generic->local).
__device__ __forceinline__ uint32_t lds_off_of(const void* p) {
  return (uint32_t)(uintptr_t)(lds_cv*)p;
}

// CDNA5 async copy: global -> LDS, 4 bytes per active lane (ASYNCcnt).
__device__ __forceinline__ void async_b32_to_lds(uint32_t lds_byte_off, const void* gaddr) {
  asm volatile("global_load_async_to_lds_b32 %0, %1, off"
               :: "v"(lds_byte_off), "v"(gaddr) : "memory");
}
#define WAIT_ASYNCCNT(n) asm volatile("s_wait_asynccnt " #n ::: "memory")

// ---------------------------------------------------------------------------
// Forward recurrence.  fvbuf = float[2][KK] double buffer in global (L2).
// bp = uint16[TT][KK] backpointers.  counter = monotonic grid barrier.
// ---------------------------------------------------------------------------
__global__ __launch_bounds__(NTHREADS, 1) void viterbi_fwd_kernel(
    const float* __restrict__ feats, const float* __restrict__ trans,
    float* __restrict__ fvbuf, uint16_t* __restrict__ bp,
    unsigned* __restrict__ counter) {
  __shared__ float fvs[KK];           // current fv, replicated per WG
  __shared__ float ftile[FR * RPW];   // feats prefetch ring (this WG's 32 rows)
  __shared__ float pval[RPW * 16];    // per-(column, prev-segment) partial max
  __shared__ int   pidx[RPW * 16];    // matching argmax (global prev index)

  const int tid  = (int)threadIdx.x;
  const int wave = tid >> 5;
  const int lane = tid & 31;
  const int wg   = (int)blockIdx.x;
  const int rbase = wg * RPW;

  // Wave (g, sseg): g = column group of 16 "next" tags, sseg = 32-wide prev
  // segment (2 WMMA tiles).  Lane L covers next column  wg*32 + g*16 + L.
  const int g    = wave >> 4;
  const int sseg = wave & 15;
  const int L    = lane & 15;
  const int h8   = (lane >> 4) << 3;          // 0 for lanes 0-15, 8 for 16-31
  const int ncol = rbase + g * 16 + L;        // next tag (column) of this lane
  const int pb   = sseg * 32;                 // prev base of this wave's segment

  // C tiles = trans^T, VGPR-resident (loaded once; 1MB across 16 WGPs).
  // C0[m][n] = trans[ncolbase+n][pb+m], C1 likewise at pb+16.  Lane layout per
  // ISA: lanes 0-15 -> rows m=0..7 in VGPRs 0..7, lanes 16-31 -> rows 8..15.
  v8f c0{}, c1{};
#pragma unroll
  for (int r = 0; r < 8; ++r) {
    c0[r] = trans[(size_t)ncol * KK + pb + h8 + r];
    c1[r] = trans[(size_t)ncol * KK + pb + 16 + h8 + r];
  }
  const v2f bones = {1.0f, 1.0f};             // B = ones (K row 0 is enough)

  // fv_0 = NEG everywhere except tag 0.
  if (tid < KK) fvs[tid] = (tid == 0) ? 0.0f : NEGV;

  const uint32_t ft_lds = lds_off_of(ftile);
  const uint32_t fv_lds = lds_off_of(fvs);

  // Prologue: async-prefetch feats rows 0..PF-1 (wave16 owns the pipeline,
  // since ASYNCcnt is per-wave).
  if (wave == 16) {
#pragma unroll
    for (int p = 0; p < PF; ++p)
      async_b32_to_lds(ft_lds + (uint32_t)(((p & (FR - 1)) * RPW + lane) * 4),
                       feats + (size_t)p * KK + rbase + lane);
  }

  for (int t = 0; t < TT; ++t) {
    // Keep the feats pipeline PF deep; clamp at the tail so exactly PF+1 are
    // outstanding and "asynccnt<=PF" always guarantees row t has landed.
    if (wave == 16) {
      int tp = t + PF; if (tp > TT - 1) tp = TT - 1;
      async_b32_to_lds(ft_lds + (uint32_t)((((t + PF) & (FR - 1)) * RPW + lane) * 4),
                       feats + (size_t)tp * KK + rbase + lane);
      WAIT_ASYNCCNT(6);  // == PF: oldest (row t) complete
    }
    __syncthreads();  // fvs (async-loaded last iter) + ftile[t] visible to all

    // A operands: column K=0 carries fv[prev] (lanes 0-15); upper lanes feed
    // K=2 and must be zero.  Exact broadcast: fv*1.0 + trans.
    float fa0 = fvs[pb + L];
    float fa1 = fvs[pb + 16 + L];
    const bool lower = (lane < 16);
    v2f a0 = {lower ? fa0 : 0.0f, 0.0f};
    v2f a1 = {lower ? fa1 : 0.0f, 0.0f};

    // D[p_local, n] = fv[pb + p_local] + trans[n][pb + p_local]
    v8f d0 = __builtin_amdgcn_wmma_f32_16x16x4_f32(false, a0, false, bones,
                                                   (short)0, c0, false, false);
    v8f d1 = __builtin_amdgcn_wmma_f32_16x16x4_f32(false, a1, false, bones,
                                                   (short)0, c1, false, false);

    // Per-lane argmax over 16 prev candidates (prev across VGPRs => no
    // shuffles).  Ascending local prev order keeps first-max-wins.
    float bestv = -3.0e38f;
    int bloc = 0;
#pragma unroll
    for (int r = 0; r < 8; ++r) {
      float s0 = d0[r];
      if (s0 > bestv) { bestv = s0; bloc = h8 + r; }
    }
#pragma unroll
    for (int r = 0; r < 8; ++r) {
      float s1 = d1[r];
      if (s1 > bestv) { bestv = s1; bloc = h8 + 16 + r; }
    }
    // Merge lane halves (same column, prev sets {0-7,16-23} vs {8-15,24-31}).
    {
      float ov = __shfl_xor(bestv, 16, 32);
      int   oi = __shfl_xor(bloc, 16, 32);
      if (ov > bestv || (ov == bestv && oi < bloc)) { bestv = ov; bloc = oi; }
    }
    // Publish this segment's partial for the lane's column.
    if (lane < 16) {
      int pslot = (g * 16 + L) * 16 + sseg;
      pval[pslot] = bestv;
      pidx[pslot] = pb + bloc;                 // global prev index
    }
    __syncthreads();

    // Final combine: wave v reduces the 16 segment-partials of column v.
    // (idx increases with segment, so tie -> lower idx matches jnp.argmax.)
    float cv = pval[wave * 16 + L];
    int   ci = pidx[wave * 16 + L];
#pragma unroll
    for (int m = 8; m >= 1; m >>= 1) {
      float ov = __shfl_xor(cv, m, 32);
      int   oi = __shfl_xor(ci, m, 32);
      if (ov > cv || (ov == cv && oi < ci)) { cv = ov; ci = oi; }
    }
    if (lane == 0) {
      float nv = cv + ftile[(t & (FR - 1)) * RPW + wave];  // + emission
      fvbuf[(size_t)((t + 1) & 1) * KK + rbase + wave] = nv;
      bp[(size_t)t * KK + rbase + wave] = (uint16_t)ci;
    }

    // ---- grid barrier: release fv writes, arrive, spin, acquire ----
    __builtin_amdgcn_fence(__ATOMIC_RELEASE, "agent");
    __syncthreads();
    if (tid == 0) {
      __hip_atomic_fetch_add(counter, 1u, __ATOMIC_RELAXED, __HIP_MEMORY_SCOPE_AGENT);
      const unsigned target = (unsigned)(t + 1) * NWG;
      while (__hip_atomic_load(counter, __ATOMIC_RELAXED, __HIP_MEMORY_SCOPE_AGENT) < target)
        __builtin_amdgcn_s_sleep(1);
    }
    __syncthreads();
    __builtin_amdgcn_fence(__ATOMIC_ACQUIRE, "agent");  // invalidate WGP$ (fvbuf reused)

    // Async-broadcast fv_{t+1} into LDS for the next step (waves 0..15).
    if (t + 1 < TT && wave < 16) {
      async_b32_to_lds(fv_lds + (uint32_t)(tid * 4),
                       fvbuf + (size_t)((t + 1) & 1) * KK + tid);
      WAIT_ASYNCCNT(0);
    }
  }
}

// ---------------------------------------------------------------------------
// Backtrack phase 1: per-chunk tag-composition maps.  Block c, thread k:
// follow bp from the chunk's last step down; rows staged through LDS.
// ---------------------------------------------------------------------------
__global__ __launch_bounds__(KK) void viterbi_chunkmap_kernel(
    const uint16_t* __restrict__ bp, uint16_t* __restrict__ chunkmap) {
  __shared__ uint16_t rowbuf[KK];
  const int c = (int)blockIdx.x;
  const int k = (int)threadIdx.x;
  int tag = k;
  for (int t = (c + 1) * LCH - 1; t >= c * LCH; --t) {
    rowbuf[k] = bp[(size_t)t * KK + k];          // coalesced 1KB row (L2-resident)
    if (t > c * LCH) __builtin_prefetch(&bp[(size_t)(t - 1) * KK + k], 0, 1);
    __syncthreads();
    tag = rowbuf[tag];                            // LDS gather
    __syncthreads();
  }
  chunkmap[(size_t)c * KK + k] = (uint16_t)tag;   // tag at time c*LCH-1
}

// ---------------------------------------------------------------------------
// Terminal scores + argmax, then sequential 256-step chunk-boundary walk.
// fvfin = fv_T (fvbuf buffer 0, since T is even).
// ---------------------------------------------------------------------------
__global__ __launch_bounds__(KK) void viterbi_term_kernel(
    const float* __restrict__ trans, const float* __restrict__ fvfin,
    const uint16_t* __restrict__ chunkmap, int* __restrict__ G,
    float* __restrict__ score_out) {
  __shared__ float sv[KK];
  __shared__ int   si[KK];
  const int k = (int)threadIdx.x;
  sv[k] = fvfin[k] + trans[(size_t)1 * KK + k];   // terminal = fv + transitions[1]
  si[k] = k;
  __syncthreads();
  for (int s = KK / 2; s >= 1; s >>= 1) {
    if (k < s) {
      float ov = sv[k + s]; int oi = si[k + s];
      if (ov > sv[k] || (ov == sv[k] && oi < si[k])) { sv[k] = ov; si[k] = oi; }
    }
    __syncthreads();
  }
  if (k == 0) {
    score_out[0] = sv[0];
    int gtag = si[0];         // best final tag = tag at time TT-1
    G[CCH - 1] = gtag;
    for (int c = CCH - 1; c >= 1; --c) {          // boundary tags, backwards
      gtag = chunkmap[(size_t)c * KK + gtag];     // tag at time c*LCH-1
      G[c - 1] = gtag;
    }
  }
}

// ---------------------------------------------------------------------------
// Backtrack phase 3: emit path; chunk per thread, 256 chunks in parallel.
// ---------------------------------------------------------------------------
__global__ __launch_bounds__(CCH) void viterbi_emit_kernel(
    const uint16_t* __restrict__ bp, const int* __restrict__ G,
    int* __restrict__ path_out) {
  const int c = (int)threadIdx.x;
  int tag = G[c];                                  // tag at time (c+1)*LCH-1
  for (int t = (c + 1) * LCH - 1; t >= c * LCH; --t) {
    path_out[t] = tag;
    tag = (int)bp[(size_t)t * KK + tag];
  }
}

// ---------------------------------------------------------------------------
extern "C" void kernel_launch(void* const* d_in, const int* in_sizes, int n_in,
                              void* d_out, int out_size, void* d_ws, size_t ws_size,
                              hipStream_t stream) {
  (void)in_sizes; (void)n_in; (void)out_size;
  const float* feats = (const float*)d_in[0];   // [TT,KK] f32
  const float* trans = (const float*)d_in[1];   // [KK,KK] f32

  float* outF = (float*)d_out;                  // out[0] = path_score (f32)
  int*   path = (int*)d_out + 1;                // out[1..TT] = best_path (i32 bits)

  // Workspace layout (needs ~64.3 MB).
  uint8_t* ws = (uint8_t*)d_ws;
  size_t off = 0;
  uint16_t* bp       = (uint16_t*)(ws + off); off += (size_t)TT * KK * sizeof(uint16_t); // 64MB
  float*    fvbuf    = (float*)(ws + off);    off += (size_t)2 * KK * sizeof(float);     // 4KB
  unsigned* counter  = (unsigned*)(ws + off); off += 256;
  uint16_t* chunkmap = (uint16_t*)(ws + off); off += (size_t)CCH * KK * sizeof(uint16_t);// 256KB
  int*      G        = (int*)(ws + off);      off += (size_t)CCH * sizeof(int);
  if (off > ws_size) { /* workspace too small; nothing safe to do */ }

  hipMemsetAsync(counter, 0, 256, stream);  // barrier counter must start at 0

  viterbi_fwd_kernel<<<NWG, NTHREADS, 0, stream>>>(feats, trans, fvbuf, bp, counter);
  viterbi_chunkmap_kernel<<<CCH, KK, 0, stream>>>(bp, chunkmap);
  viterbi_term_kernel<<<1, KK, 0, stream>>>(trans, fvbuf /* buffer 0 = fv_T */,
                                            chunkmap, G, outF);
  viterbi_emit_kernel<<<1, CCH, 0, stream>>>(bp, G, path);
}